// IxformerMoEInitRoutingDynamicQuant_80908593922366
// MI455X (gfx1250) — compile-verified
//
#include <hip/hip_runtime.h>
#include <hip/hip_bf16.h>
#include <stdint.h>

// Problem constants (match reference)
#define NUM_EXPERTS 64
#define TOPK        2
#define DIM         2048
#define QTHREADS    256          // DIM / 8 elements per thread
#define ROWS_PER_BLOCK 8         // rows pipelined per block (smooth staged once)

typedef float vfloat4 __attribute__((ext_vector_type(4)));

// ---------------------------------------------------------------------------
// CDNA5 async global->LDS helpers (ASYNCcnt-tracked, CDNA5 ISA)
// ---------------------------------------------------------------------------
__device__ __forceinline__ unsigned lds_off32(const void* p) {
  // Flat LDS aperture: addr[31:0] == wave-relative LDS byte offset (ISA 10.2)
  return (unsigned)(unsigned long long)p;
}

__device__ __forceinline__ void async_copy_b128(unsigned lds_byte_off, const void* gaddr) {
  // GLOBAL_LOAD_ASYNC_TO_LDS_B128: per-lane 16B memory -> LDS, tracked by ASYNCcnt
  asm volatile("global_load_async_to_lds_b128 %0, %1, off"
               :: "v"(lds_byte_off), "v"(gaddr)
               : "memory");
}

__device__ __forceinline__ void wait_async_zero() {
  asm volatile("s_wait_asynccnt 0" ::: "memory");
}

// ---------------------------------------------------------------------------
// Kernel 1: per-expert histogram + exclusive prefix (single block)
// ---------------------------------------------------------------------------
__global__ __launch_bounds__(1024)
void count_experts_kernel(const int* __restrict__ experts, int n,
                          int* __restrict__ offsets, float* __restrict__ out_sizes) {
  __shared__ int cnt[NUM_EXPERTS];
  if (threadIdx.x < NUM_EXPERTS) cnt[threadIdx.x] = 0;
  __syncthreads();
  for (int i = threadIdx.x; i < n; i += blockDim.x) {
    atomicAdd(&cnt[experts[i]], 1);
  }
  __syncthreads();
  if (threadIdx.x == 0) {
    int run = 0;
    for (int e = 0; e < NUM_EXPERTS; ++e) { offsets[e] = run; run += cnt[e]; }
    offsets[NUM_EXPERTS] = run;
  }
  if (threadIdx.x < NUM_EXPERTS) out_sizes[threadIdx.x] = (float)cnt[threadIdx.x];
}

// ---------------------------------------------------------------------------
// Kernel 2: stable bucket gather. One block per expert; wave32 ballot prefix
// preserves original slot order (== stable argsort of expert ids).
// ---------------------------------------------------------------------------
__global__ __launch_bounds__(256)
void gather_expert_kernel(const int* __restrict__ experts,
                          const int* __restrict__ offsets,
                          int*   __restrict__ sorted_src,
                          float* __restrict__ out_sorted,
                          float* __restrict__ out_s2d, int n) {
  const int e    = blockIdx.x;
  const int lane = threadIdx.x & 31;
  const int wave = threadIdx.x >> 5;   // 8 waves of 32 (wave32)
  __shared__ int waveCnt[8];
  __shared__ int sBase;
  if (threadIdx.x == 0) sBase = offsets[e];
  __syncthreads();

  for (int chunk = 0; chunk < n; chunk += (int)blockDim.x) {
    const int i = chunk + threadIdx.x;
    const bool m = (i < n) && (experts[i] == e);
    const unsigned mask = (unsigned)__ballot(m);   // wave32: low 32 bits
    if (lane == 0) waveCnt[wave] = __popc(mask);
    __syncthreads();
    int wavePre = 0, total = 0;
#pragma unroll
    for (int w = 0; w < 8; ++w) {
      const int c = waveCnt[w];
      if (w < wave) wavePre += c;
      total += c;
    }
    if (m) {
      const int rank = __popc(mask & ((1u << lane) - 1u));
      const int dst  = sBase + wavePre + rank;
      sorted_src[dst] = i;
      out_sorted[dst] = (float)i;
      out_s2d[i]      = (float)dst;
    }
    __syncthreads();
    if (threadIdx.x == 0) sBase += total;
    __syncthreads();
  }
}

// ---------------------------------------------------------------------------
// Kernel 3: gates passthrough (non-temporal both ways; never re-read)
// ---------------------------------------------------------------------------
__global__ void copy_gates_kernel(const float* __restrict__ g, float* __restrict__ o, int n) {
  for (int i = blockIdx.x * blockDim.x + threadIdx.x; i < n; i += gridDim.x * blockDim.x) {
    float v = __builtin_nontemporal_load(g + i);
    __builtin_nontemporal_store(v, o + i);
  }
}

// ---------------------------------------------------------------------------
// Kernel 4: gather + smooth + per-row absmax + int8 quantize (stored as float).
// ROWS_PER_BLOCK rows per block; smooth_scale staged in LDS once; row tiles
// double-buffered with async b128 copies so the fetch of row i+1 hides behind
// the quantize of row i. q output stored non-temporal to keep x L2-resident.
// ---------------------------------------------------------------------------
__global__ __launch_bounds__(QTHREADS)
void quant_rows_kernel(const float* __restrict__ x,
                       const float* __restrict__ smooth,
                       const int*   __restrict__ sorted_src,
                       float* __restrict__ out_q,
                       float* __restrict__ out_qscale, int nExp) {
  __shared__ __align__(16) float ldsSm[DIM];
  __shared__ __align__(16) float ldsBuf[2][DIM];
  __shared__ float ldsRed[8];

  const int tid  = threadIdx.x;
  const int lane = tid & 31;
  const int wave = tid >> 5;
  const int base = blockIdx.x * ROWS_PER_BLOCK;

  const unsigned smOff = lds_off32(&ldsSm[tid * 4]);
  const unsigned bOff[2] = { lds_off32(&ldsBuf[0][tid * 4]),
                             lds_off32(&ldsBuf[1][tid * 4]) };

  // Stage smooth_scale once per block (8KB), plus row 0 into buffer 0.
  async_copy_b128(smOff,        smooth + tid * 4);
  async_copy_b128(smOff + 4096, smooth + tid * 4 + 1024);

  int dst0 = base; if (dst0 >= nExp) dst0 = nExp - 1;
  {
    const float* r0 = x + (size_t)(sorted_src[dst0] / TOPK) * DIM;
    async_copy_b128(bOff[0],        r0 + tid * 4);
    async_copy_b128(bOff[0] + 4096, r0 + tid * 4 + 1024);
  }
  wait_async_zero();
  __syncthreads();          // smooth + row0 visible to all waves

  for (int i = 0; i < ROWS_PER_BLOCK; ++i) {
    int dst = base + i; if (dst >= nExp) dst = nExp - 1;

    // Issue async fetch of the next row into the other buffer. Safe: the
    // end-of-previous-iteration barrier guarantees all reads of it are done.
    if (i + 1 < ROWS_PER_BLOCK) {
      int dstn = base + i + 1; if (dstn >= nExp) dstn = nExp - 1;
      const float* rn = x + (size_t)(sorted_src[dstn] / TOPK) * DIM;
      const unsigned nb = bOff[(i + 1) & 1];
      async_copy_b128(nb,        rn + tid * 4);
      async_copy_b128(nb + 4096, rn + tid * 4 + 1024);
    }

    // ---- compute current row from LDS ----
    const float* cur = ldsBuf[i & 1];
    const vfloat4 a0 = *(const vfloat4*)(cur + tid * 8);
    const vfloat4 a1 = *(const vfloat4*)(cur + tid * 8 + 4);
    const vfloat4 s0 = *(const vfloat4*)(ldsSm + tid * 8);
    const vfloat4 s1 = *(const vfloat4*)(ldsSm + tid * 8 + 4);

    float v[8];
    v[0] = a0.x * s0.x; v[1] = a0.y * s0.y; v[2] = a0.z * s0.z; v[3] = a0.w * s0.w;
    v[4] = a1.x * s1.x; v[5] = a1.y * s1.y; v[6] = a1.z * s1.z; v[7] = a1.w * s1.w;

    float m = 0.0f;
#pragma unroll
    for (int k = 0; k < 8; ++k) m = fmaxf(m, fabsf(v[k]));
#pragma unroll
    for (int off = 16; off > 0; off >>= 1) m = fmaxf(m, __shfl_xor(m, off, 32));
    if (lane == 0) ldsRed[wave] = m;
    __syncthreads();
    float amax = 0.0f;
#pragma unroll
    for (int w = 0; w < 8; ++w) amax = fmaxf(amax, ldsRed[w]);

    const float qs   = amax * (1.0f / 127.0f);
    const float safe = (qs == 0.0f) ? 1.0f : qs;

    float q[8];
#pragma unroll
    for (int k = 0; k < 8; ++k) {
      float r = rintf(v[k] / safe);          // round-half-even == jnp.round
      q[k] = fminf(fmaxf(r, -127.0f), 127.0f);
    }

    float* orow = out_q + (size_t)dst * DIM + tid * 8;
    vfloat4 q0 = { q[0], q[1], q[2], q[3] };
    vfloat4 q1 = { q[4], q[5], q[6], q[7] };
    __builtin_nontemporal_store(q0, (vfloat4*)orow);        // TH=NT stream-out
    __builtin_nontemporal_store(q1, (vfloat4*)(orow + 4));
    if (tid == 0) out_qscale[dst] = qs;

    // Drain this wave's async ops (next row landed; fetch latency was hidden
    // behind the quantize above), then one barrier certifies both "all reads
    // of the current buffer done" and "next buffer data visible".
    wait_async_zero();
    __syncthreads();
  }
}

// ---------------------------------------------------------------------------
// Launcher
// ---------------------------------------------------------------------------
extern "C" void kernel_launch(void* const* d_in, const int* in_sizes, int n_in,
                              void* d_out, int out_size, void* d_ws, size_t ws_size,
                              hipStream_t stream) {
  const float* hidden  = (const float*)d_in[0];   // [T, D] flat
  const float* gates   = (const float*)d_in[1];   // [T*K]
  const int*   experts = (const int*)  d_in[2];   // [T*K]
  const float* smooth  = (const float*)d_in[3];   // [D]

  const int nExp = in_sizes[1];                   // T*K == 32768
  const size_t D = (size_t)in_sizes[3];           // 2048

  float* out        = (float*)d_out;
  float* OUT_Q      = out;                        // nExp*D
  float* OUT_GATES  = OUT_Q + (size_t)nExp * D;   // nExp
  float* OUT_SORTED = OUT_GATES + nExp;           // nExp
  float* OUT_S2D    = OUT_SORTED + nExp;          // nExp
  float* OUT_SIZES  = OUT_S2D + nExp;             // NUM_EXPERTS
  float* OUT_QSCALE = OUT_SIZES + NUM_EXPERTS;    // nExp

  int* offsets    = (int*)d_ws;                   // [NUM_EXPERTS+1]
  int* sorted_src = offsets + 128;                // [nExp]

  count_experts_kernel<<<1, 1024, 0, stream>>>(experts, nExp, offsets, OUT_SIZES);
  gather_expert_kernel<<<NUM_EXPERTS, 256, 0, stream>>>(
      experts, offsets, sorted_src, OUT_SORTED, OUT_S2D, nExp);
  copy_gates_kernel<<<128, 256, 0, stream>>>(gates, OUT_GATES, nExp);

  const int qBlocks = (nExp + ROWS_PER_BLOCK - 1) / ROWS_PER_BLOCK;
  quant_rows_kernel<<<qBlocks, QTHREADS, 0, stream>>>(
      hidden, smooth, sorted_src, OUT_Q, OUT_QSCALE, nExp);
}